// DEDistMult_18786186953558
// MI455X (gfx1250) — compile-verified
//
#include <hip/hip_runtime.h>

typedef __attribute__((ext_vector_type(2))) float v2f;
typedef __attribute__((ext_vector_type(8))) float v8f;

#define SDIM 128
#define TDIM 64
#define KDIM 192
#define TPB  16            // triples per block (one wave32 per block)
#define LDS_STRIDE 196     // 196 % 64 == 4 -> conflict-free column reads; *4B is 8B-aligned

__device__ __forceinline__ float2 ld2(const float* __restrict__ p) {
    return *(const float2*)p;
}

// temporal embedding for one entity, 2 consecutive dims (this lane's slice)
__device__ __forceinline__ float2 temb2(size_t b, float yv, float mv, float dv,
                                        const float* __restrict__ yf, const float* __restrict__ mf,
                                        const float* __restrict__ df, const float* __restrict__ yp,
                                        const float* __restrict__ mp, const float* __restrict__ dp,
                                        const float* __restrict__ ya, const float* __restrict__ ma,
                                        const float* __restrict__ da) {
    float2 f, p, a, r;
    f = ld2(yf + b); p = ld2(yp + b); a = ld2(ya + b);
    r.x = a.x * __sinf(f.x * yv + p.x);
    r.y = a.y * __sinf(f.y * yv + p.y);
    f = ld2(mf + b); p = ld2(mp + b); a = ld2(ma + b);
    r.x += a.x * __sinf(f.x * mv + p.x);
    r.y += a.y * __sinf(f.y * mv + p.y);
    f = ld2(df + b); p = ld2(dp + b); a = ld2(da + b);
    r.x += a.x * __sinf(f.x * dv + p.x);
    r.y += a.y * __sinf(f.y * dv + p.y);
    return r;
}

__global__ __launch_bounds__(32)
void dedistmult_kernel(const int* __restrict__ s, const int* __restrict__ r,
                       const int* __restrict__ o, const float* __restrict__ y,
                       const float* __restrict__ m, const float* __restrict__ d,
                       const float* __restrict__ e_emb, const float* __restrict__ rel_emb,
                       const float* __restrict__ y_frq, const float* __restrict__ m_frq,
                       const float* __restrict__ d_frq, const float* __restrict__ y_phi,
                       const float* __restrict__ m_phi, const float* __restrict__ d_phi,
                       const float* __restrict__ y_amp, const float* __restrict__ m_amp,
                       const float* __restrict__ d_amp, float* __restrict__ out)
{
    __shared__ float P[TPB * LDS_STRIDE];   // P[t][k] = s_emb[k]*rel[k]*o_emb[k]
    const int lane = threadIdx.x;           // 0..31
    const int tb   = blockIdx.x * TPB;

    // ---------------- Phase 1: coalesced gather + fuse into LDS ----------------
    #pragma unroll 1
    for (int tt = 0; tt < TPB; ++tt) {
        const int t  = tb + tt;
        const int si = s[t], ri = r[t], oi = o[t];
        const float yv = y[t], mv = m[t], dv = d[t];

        // static part: dims [4*lane .. 4*lane+3]  (b128 loads)
        const float4 es = *(const float4*)(e_emb  + (size_t)si * SDIM + lane * 4);
        const float4 eo = *(const float4*)(e_emb  + (size_t)oi * SDIM + lane * 4);
        const float4 rr = *(const float4*)(rel_emb + (size_t)ri * KDIM + lane * 4);
        float4 p4;
        p4.x = es.x * rr.x * eo.x;
        p4.y = es.y * rr.y * eo.y;
        p4.z = es.z * rr.z * eo.z;
        p4.w = es.w * rr.w * eo.w;
        *(float4*)(&P[tt * LDS_STRIDE + lane * 4]) = p4;   // ds_store_b128

        // temporal part: dims [128 + 2*lane .. +1]  (b64 loads)
        const size_t bs = (size_t)si * TDIM + lane * 2;
        const size_t bo = (size_t)oi * TDIM + lane * 2;
        const float2 ts = temb2(bs, yv, mv, dv, y_frq, m_frq, d_frq,
                                y_phi, m_phi, d_phi, y_amp, m_amp, d_amp);
        const float2 to = temb2(bo, yv, mv, dv, y_frq, m_frq, d_frq,
                                y_phi, m_phi, d_phi, y_amp, m_amp, d_amp);
        const float2 r2 = ld2(rel_emb + (size_t)ri * KDIM + SDIM + lane * 2);
        float2 pt;
        pt.x = ts.x * r2.x * to.x;
        pt.y = ts.y * r2.y * to.y;
        *(float2*)(&P[tt * LDS_STRIDE + SDIM + lane * 2]) = pt;  // ds_store_b64
    }
    // Single-wave workgroup: LDS ops from this wave are in order; no barrier needed.

    // ---------------- Phase 2: 192-dim row reduction on the matrix pipe --------
    // score(16 rows) = P(16x192) x ones(192x16), K consumed 4 at a time with
    // V_WMMA_F32_16X16X4_F32.  A-layout (16x4 f32): lane L holds row M = L&15,
    // VGPR v = element K = 2*(L>>4) + v  -> one ds_load_b64 per chunk.
    const int tl   = lane & 15;
    const int half = lane >> 4;
    const float* rowp = &P[tl * LDS_STRIDE + 2 * half];

    v2f bones = {1.0f, 1.0f};                       // B = all-ones, layout-invariant
    v8f acc0 = {0,0,0,0,0,0,0,0};
    v8f acc1 = {0,0,0,0,0,0,0,0};

    #pragma unroll
    for (int k0 = 0; k0 < KDIM; k0 += 8) {
        v2f a0 = *(const v2f*)(rowp + k0);          // dims k0   + 2*half + {0,1}
        v2f a1 = *(const v2f*)(rowp + k0 + 4);      // dims k0+4 + 2*half + {0,1}
        acc0 = __builtin_amdgcn_wmma_f32_16x16x4_f32(
                   false, a0, false, bones, (short)0, acc0, false, false);
        acc1 = __builtin_amdgcn_wmma_f32_16x16x4_f32(
                   false, a1, false, bones, (short)0, acc1, false, false);
    }

    // D layout: VGPR v, lanes 0-15 hold M=v (N=lane), lanes 16-31 hold M=v+8.
    // Row-sums are replicated across all N columns; lane with N==0 in each half
    // owns scores tb + half*8 + v in its 8 accumulator elements.
    if (tl == 0) {
        const int base = tb + half * 8;
        #pragma unroll
        for (int v = 0; v < 8; ++v)
            out[base + v] = acc0[v] + acc1[v];
    }
}

extern "C" void kernel_launch(void* const* d_in, const int* in_sizes, int n_in,
                              void* d_out, int out_size, void* d_ws, size_t ws_size,
                              hipStream_t stream) {
    const int*   s       = (const int*)d_in[0];
    const int*   r       = (const int*)d_in[1];
    const int*   o       = (const int*)d_in[2];
    const float* y       = (const float*)d_in[3];
    const float* m       = (const float*)d_in[4];
    const float* d       = (const float*)d_in[5];
    const float* e_emb   = (const float*)d_in[6];
    const float* rel_emb = (const float*)d_in[7];
    const float* y_frq   = (const float*)d_in[8];
    const float* m_frq   = (const float*)d_in[9];
    const float* d_frq   = (const float*)d_in[10];
    const float* y_phi   = (const float*)d_in[11];
    const float* m_phi   = (const float*)d_in[12];
    const float* d_phi   = (const float*)d_in[13];
    const float* y_amp   = (const float*)d_in[14];
    const float* m_amp   = (const float*)d_in[15];
    const float* d_amp   = (const float*)d_in[16];
    float* out = (float*)d_out;

    const int B = in_sizes[0];          // 131072, divisible by TPB
    dim3 grid(B / TPB), block(32);
    dedistmult_kernel<<<grid, block, 0, stream>>>(
        s, r, o, y, m, d, e_emb, rel_emb,
        y_frq, m_frq, d_frq, y_phi, m_phi, d_phi, y_amp, m_amp, d_amp, out);
}